// CaptionModel_42417097017011
// MI455X (gfx1250) — compile-verified
//
#include <hip/hip_runtime.h>
#include <hip/hip_bf16.h>

typedef __attribute__((ext_vector_type(2))) float v2f;
typedef __attribute__((ext_vector_type(8))) float v8f;

#define BB 10          // beams
#define VV 50257       // vocab
#define VP 50272       // vocab padded to 16
#define DD 1024        // hidden
#define TT 20          // steps
#define NTILES ((VV + 15) / 16)

// ---------------------------------------------------------------------------
// init: h_cur[16][1024] from h0 (pad rows 10..15 with 0), zero beam state
// ---------------------------------------------------------------------------
__global__ void k_init(const float* __restrict__ h0, float* __restrict__ hcur,
                       int* __restrict__ bseq, float* __restrict__ blp,
                       float* __restrict__ bsum) {
  int idx = blockIdx.x * blockDim.x + threadIdx.x;   // 0..16383
  int row = idx >> 10, col = idx & 1023;
  if (row < 16) hcur[idx] = (row < BB) ? h0[row * DD + col] : 0.0f;
  if (idx < TT * BB) { bseq[idx] = 0; blp[idx] = 0.0f; }
  if (idx < BB) bsum[idx] = 0.0f;
}

// ---------------------------------------------------------------------------
// per-beam top-10 over V.  t==0 reads logprobs0 directly; t>0 reads raw
// logits and normalizes with (max, log-sum-exp).  UNK (col V-1) -= 1000.
// ---------------------------------------------------------------------------
__global__ __launch_bounds__(256) void k_topk(
    const float* __restrict__ lp0, const float* __restrict__ logits,
    const float* __restrict__ maxv, const float* __restrict__ lsum,
    float* __restrict__ topv, int* __restrict__ topi, int use_input) {
  int b = blockIdx.x, tid = threadIdx.x;
  __shared__ float sv[256 * 10];
  __shared__ int   si[256 * 10];
  __shared__ float mv[32 * 10];
  __shared__ int   mi[32 * 10];

  float lv[10]; int li[10];
#pragma unroll
  for (int j = 0; j < 10; ++j) { lv[j] = -3.0e38f; li[j] = -1; }

  float mm = use_input ? 0.0f : maxv[b];
  float ll = use_input ? 0.0f : lsum[b];
  const float* row = use_input ? (lp0 + (size_t)b * VV) : (logits + (size_t)b * VP);

  for (int i = tid; i < VV; i += 256) {
    float v = row[i] - mm - ll;
    if (i == VV - 1) v -= 1000.0f;      // suppress UNK
    if (v > lv[9]) {
      int ci = i;
#pragma unroll
      for (int p = 0; p < 10; ++p) {
        if (v > lv[p]) { float tv = lv[p]; int ti = li[p];
                         lv[p] = v; li[p] = ci; v = tv; ci = ti; }
      }
    }
  }
#pragma unroll
  for (int j = 0; j < 10; ++j) { sv[tid * 10 + j] = lv[j]; si[tid * 10 + j] = li[j]; }
  __syncthreads();

  if (tid < 32) {           // merge 8 sorted lists -> one sorted 10
    int hp[8];
#pragma unroll
    for (int l = 0; l < 8; ++l) hp[l] = 0;
    for (int r = 0; r < 10; ++r) {
      float best = -3.0e38f; int bl = 0;
      for (int l = 0; l < 8; ++l) {
        if (hp[l] < 10) {
          float v = sv[(tid + l * 32) * 10 + hp[l]];
          if (v > best) { best = v; bl = l; }
        }
      }
      mv[tid * 10 + r] = best;
      mi[tid * 10 + r] = si[(tid + bl * 32) * 10 + hp[bl]];
      hp[bl]++;
    }
  }
  __syncthreads();

  if (tid == 0) {           // merge 32 sorted lists -> final top-10
    int hp[32];
    for (int l = 0; l < 32; ++l) hp[l] = 0;
    for (int r = 0; r < 10; ++r) {
      float best = -3.0e38f; int bl = 0;
      for (int l = 0; l < 32; ++l) {
        if (hp[l] < 10) {
          float v = mv[l * 10 + hp[l]];
          if (v > best) { best = v; bl = l; }
        }
      }
      topv[b * 10 + r] = best;
      topi[b * 10 + r] = mi[bl * 10 + hp[bl]];
      hp[bl]++;
    }
  }
}

// ---------------------------------------------------------------------------
// candidate selection + beam reorder (tiny, single block)
// ---------------------------------------------------------------------------
__global__ void k_select(const float* __restrict__ topv, const int* __restrict__ topi,
                         float* __restrict__ bsum, int* __restrict__ bseq,
                         float* __restrict__ blp, int* __restrict__ toks,
                         int* __restrict__ qsel, int t) {
  __shared__ int   s_seq[TT * BB];
  __shared__ float s_lp[TT * BB];
  __shared__ int   s_q[BB];
  __shared__ int   s_tok[BB];
  __shared__ float s_loc[BB];
  int tid = threadIdx.x;

  for (int i = tid; i < TT * BB; i += 64) { s_seq[i] = bseq[i]; s_lp[i] = blp[i]; }
  __syncthreads();

  if (tid == 0) {
    float cand[100]; bool used[100];
    for (int i = 0; i < 100; ++i) {
      int bi = i / 10;
      cand[i] = bsum[bi] + topv[i];
      if (t == 0 && bi > 0) cand[i] = -1.0e10f;
      used[i] = false;
    }
    for (int r = 0; r < BB; ++r) {
      float best = -3.0e38f; int bj = 0;
      for (int j = 0; j < 100; ++j)
        if (!used[j] && cand[j] > best) { best = cand[j]; bj = j; }
      used[bj] = true;
      int q = bj / 10;
      int tok = topi[bj];
      s_q[r] = q; s_tok[r] = tok; s_loc[r] = topv[bj];
      bsum[r] = (tok == 0) ? -1000.0f : best;
      toks[r] = tok; qsel[r] = q;
    }
  }
  __syncthreads();

  for (int idx = tid; idx < TT * BB; idx += 64) {
    int row = idx / BB, r = idx % BB;
    bseq[idx] = s_seq[row * BB + s_q[r]];
    blp[idx]  = s_lp[row * BB + s_q[r]];
  }
  __syncthreads();
  if (tid < BB) { bseq[t * BB + tid] = s_tok[tid]; blp[t * BB + tid] = s_loc[tid]; }
}

// ---------------------------------------------------------------------------
// gather emb[tokens] and h[q] into padded [16][1024] tiles
// ---------------------------------------------------------------------------
__global__ void k_gather(const float* __restrict__ emb, const float* __restrict__ hcur,
                         const int* __restrict__ toks, const int* __restrict__ qsel,
                         float* __restrict__ xg, float* __restrict__ hg) {
  int idx = blockIdx.x * blockDim.x + threadIdx.x;   // 0..16383
  int row = idx >> 10, col = idx & 1023;
  if (row < 16) {
    xg[idx] = (row < BB) ? emb[(size_t)toks[row] * DD + col] : 0.0f;
    hg[idx] = (row < BB) ? hcur[qsel[row] * DD + col] : 0.0f;
  }
}

// ---------------------------------------------------------------------------
// RNN step: h = tanh(xg @ wx + hg @ wh), one wave per 16x16 output tile,
// V_WMMA_F32_16X16X4_F32 over K=1024
// ---------------------------------------------------------------------------
__global__ __launch_bounds__(32) void k_rnn(
    const float* __restrict__ xg, const float* __restrict__ hg,
    const float* __restrict__ wx, const float* __restrict__ wh,
    float* __restrict__ hcur) {
  int n0 = blockIdx.x * 16;
  int lane = threadIdx.x;
  int half = lane >> 4;          // 0: K pair {k,k+1}, 1: {k+2,k+3}
  int l16 = lane & 15;
  v8f acc = {};
  for (int k = 0; k < DD; k += 4) {
    int ka = k + half * 2;
    v2f a1, b1, a2, b2;
    a1.x = xg[l16 * DD + ka];       a1.y = xg[l16 * DD + ka + 1];
    b1.x = wx[(size_t)ka * DD + n0 + l16];
    b1.y = wx[(size_t)(ka + 1) * DD + n0 + l16];
    acc = __builtin_amdgcn_wmma_f32_16x16x4_f32(false, a1, false, b1,
                                                (short)0, acc, false, false);
    a2.x = hg[l16 * DD + ka];       a2.y = hg[l16 * DD + ka + 1];
    b2.x = wh[(size_t)ka * DD + n0 + l16];
    b2.y = wh[(size_t)(ka + 1) * DD + n0 + l16];
    acc = __builtin_amdgcn_wmma_f32_16x16x4_f32(false, a2, false, b2,
                                                (short)0, acc, false, false);
  }
#pragma unroll
  for (int v = 0; v < 8; ++v) {
    int m = v + half * 8;
    hcur[m * DD + n0 + l16] = tanhf(acc[v]);   // pad rows stay 0 (tanh(0)=0)
  }
}

// ---------------------------------------------------------------------------
// logits = h @ wo : one wave per 16-column tile of V, streams wo once.
// ---------------------------------------------------------------------------
__global__ __launch_bounds__(32) void k_logits(
    const float* __restrict__ hcur, const float* __restrict__ wo,
    float* __restrict__ logits) {
  int n0 = blockIdx.x * 16;
  int lane = threadIdx.x;
  int half = lane >> 4;
  int l16 = lane & 15;
  int col = n0 + l16;
  bool colok = col < VV;
  v8f acc = {};
  for (int k = 0; k < DD; k += 4) {
    int ka = k + half * 2;
    v2f a, b;
    a.x = hcur[l16 * DD + ka];
    a.y = hcur[l16 * DD + ka + 1];
    b.x = colok ? wo[(size_t)ka * VV + col] : 0.0f;
    b.y = colok ? wo[(size_t)(ka + 1) * VV + col] : 0.0f;
    acc = __builtin_amdgcn_wmma_f32_16x16x4_f32(false, a, false, b,
                                                (short)0, acc, false, false);
  }
#pragma unroll
  for (int v = 0; v < 8; ++v) {
    int m = v + half * 8;
    if (m < BB && colok) logits[(size_t)m * VP + col] = acc[v];
  }
}

// ---------------------------------------------------------------------------
// per-beam log-softmax statistics (max, log-sum-exp)
// ---------------------------------------------------------------------------
__global__ __launch_bounds__(256) void k_stats(const float* __restrict__ logits,
                                               float* __restrict__ maxv,
                                               float* __restrict__ lsum) {
  int b = blockIdx.x, tid = threadIdx.x;
  __shared__ float red[256];
  const float* row = logits + (size_t)b * VP;
  float m = -3.0e38f;
  for (int i = tid; i < VV; i += 256) m = fmaxf(m, row[i]);
  red[tid] = m; __syncthreads();
  for (int s = 128; s > 0; s >>= 1) {
    if (tid < s) red[tid] = fmaxf(red[tid], red[tid + s]);
    __syncthreads();
  }
  m = red[0]; __syncthreads();
  float sum = 0.0f;
  for (int i = tid; i < VV; i += 256) sum += __expf(row[i] - m);
  red[tid] = sum; __syncthreads();
  for (int s = 128; s > 0; s >>= 1) {
    if (tid < s) red[tid] += red[tid + s];
    __syncthreads();
  }
  if (tid == 0) { maxv[b] = m; lsum[b] = logf(red[0]); }
}

// ---------------------------------------------------------------------------
// emit outputs: [T*B] beam_seq (as float), [T*B] beam_lp, [B] beam_sum
// ---------------------------------------------------------------------------
__global__ void k_out(const int* __restrict__ bseq, const float* __restrict__ blp,
                      const float* __restrict__ bsum, float* __restrict__ out) {
  int i = threadIdx.x;
  if (i < TT * BB) out[i] = (float)bseq[i];
  if (i < TT * BB) out[TT * BB + i] = blp[i];
  if (i < BB) out[2 * TT * BB + i] = bsum[i];
}

extern "C" void kernel_launch(void* const* d_in, const int* in_sizes, int n_in,
                              void* d_out, int out_size, void* d_ws, size_t ws_size,
                              hipStream_t stream) {
  const float* lp0 = (const float*)d_in[0];
  const float* h0  = (const float*)d_in[1];
  const float* emb = (const float*)d_in[2];
  const float* wx  = (const float*)d_in[3];
  const float* wh  = (const float*)d_in[4];
  const float* wo  = (const float*)d_in[5];
  // d_in[6] = seq_length (device scalar); T fixed at 20 per setup.

  size_t off = 0;
  auto alloc = [&](size_t nbytes) -> void* {
    void* p = (char*)d_ws + off;
    off += (nbytes + 255) & ~(size_t)255;
    return p;
  };
  float* logits = (float*)alloc((size_t)BB * VP * sizeof(float));
  float* hcur   = (float*)alloc(16 * DD * sizeof(float));
  float* xg     = (float*)alloc(16 * DD * sizeof(float));
  float* hg     = (float*)alloc(16 * DD * sizeof(float));
  float* topv   = (float*)alloc(100 * sizeof(float));
  int*   topi   = (int*)  alloc(100 * sizeof(int));
  int*   bseq   = (int*)  alloc(TT * BB * sizeof(int));
  float* blp    = (float*)alloc(TT * BB * sizeof(float));
  float* bsum   = (float*)alloc(16 * sizeof(float));
  int*   toks   = (int*)  alloc(16 * sizeof(int));
  int*   qsel   = (int*)  alloc(16 * sizeof(int));
  float* maxv   = (float*)alloc(16 * sizeof(float));
  float* lsum   = (float*)alloc(16 * sizeof(float));

  k_init<<<64, 256, 0, stream>>>(h0, hcur, bseq, blp, bsum);

  for (int t = 0; t < TT; ++t) {
    k_topk<<<BB, 256, 0, stream>>>(lp0, logits, maxv, lsum, topv, topi,
                                   (t == 0) ? 1 : 0);
    k_select<<<1, 64, 0, stream>>>(topv, topi, bsum, bseq, blp, toks, qsel, t);
    k_gather<<<64, 256, 0, stream>>>(emb, hcur, toks, qsel, xg, hg);
    k_rnn<<<DD / 16, 32, 0, stream>>>(xg, hg, wx, wh, hcur);
    if (t + 1 < TT) {
      k_logits<<<NTILES, 32, 0, stream>>>(hcur, wo, logits);
      k_stats<<<BB, 256, 0, stream>>>(logits, maxv, lsum);
    }
  }
  k_out<<<1, 256, 0, stream>>>(bseq, blp, bsum, (float*)d_out);
}